// RelationGCNLayer_2662879724148
// MI455X (gfx1250) — compile-verified
//
#include <hip/hip_runtime.h>
#include <math.h>

typedef __attribute__((ext_vector_type(2))) float v2f;
typedef __attribute__((ext_vector_type(8))) float v8f;

#define D 128  // D_IN == D_OUT == 128

// ---------------------------------------------------------------------------
// Kernel 1: x_trans = x @ W_lin^T   (fp32 WMMA 16x16x4)
//   C[m,n] = sum_k x[m,k] * W[n,k]      (W stored [out,in] -> B[k][n] = W[n][k])
// Block: 256 threads = 8 waves. Block owns a 16-row M tile; wave w owns the
// 16-col N tile [16w, 16w+16). x tile staged in LDS with coalesced loads.
// A VGPR layout (32b, 16x4): lane%16 = M, K = v + 2*(lane/16)
// B VGPR layout (32b, 4x16): lane%16 = N, K = v + 2*(lane/16)
// C/D layout   (32b, 16x16): VGPR r -> M = r + 8*(lane/16), N = lane%16
// ---------------------------------------------------------------------------
__global__ __launch_bounds__(256)
void rgcn_lin_wmma(const float* __restrict__ x, const float* __restrict__ W,
                   float* __restrict__ xt, int nrows) {
    __shared__ float xs[16 * D];
    const int row0 = blockIdx.x * 16;

    // coalesced 16xD tile load (each thread 8 contiguous-strided floats)
    for (int i = threadIdx.x; i < 16 * D; i += 256) {
        int r = row0 + (i >> 7);
        xs[i] = (r < nrows) ? x[(size_t)r * D + (i & 127)] : 0.0f;
    }
    __syncthreads();

    const int wave = threadIdx.x >> 5;   // N-tile index 0..7
    const int lane = threadIdx.x & 31;
    const int half = lane >> 4;
    const int mn   = lane & 15;          // M for A, N for B/C
    const float* wrow = W + (size_t)(wave * 16 + mn) * D;  // row n of W_lin

    v8f c = {};
#pragma unroll 8
    for (int kb = 0; kb < D; kb += 4) {
        const int k = kb + 2 * half;
        v2f a; a.x = xs[mn * D + k];  a.y = xs[mn * D + k + 1];
        v2f b; b.x = wrow[k];         b.y = wrow[k + 1];
        // 8 args: (neg_a, A, neg_b, B, c_mod, C, reuse_a, reuse_b)
        c = __builtin_amdgcn_wmma_f32_16x16x4_f32(false, a, false, b,
                                                  (short)0, c, false, false);
    }

    const int col = wave * 16 + mn;
    float* base = xt + (size_t)(row0 + 8 * half) * D + col;
    if (row0 + 16 <= nrows) {
        // uniform (scalar) condition: whole tile in bounds -> straight-line
        // unguarded stores, no per-row exec-mask juggling
#pragma unroll
        for (int r = 0; r < 8; ++r)
            base[(size_t)r * D] = c[r];
    } else {
        // ragged final tile only
#pragma unroll
        for (int r = 0; r < 8; ++r) {
            int row = row0 + r + 8 * half;
            if (row < nrows) xt[(size_t)row * D + col] = c[r];
        }
    }
}

// ---------------------------------------------------------------------------
// Kernel 2: one wave32 per edge.
//   attn = sigmoid( dot(x[src]+rel_emb[type], W_attn) )
//   out[tgt] += attn * x_trans[src]        (global_atomic_add_f32)
// Each lane owns 4 consecutive floats -> every row access is a contiguous,
// fully coalesced 512B transaction per wave.
// ---------------------------------------------------------------------------
__global__ __launch_bounds__(256)
void rgcn_edge(const float* __restrict__ x, const int* __restrict__ ei,
               const int* __restrict__ et, const float* __restrict__ rel,
               const float* __restrict__ wattn, const float* __restrict__ xt,
               float* __restrict__ out, int E) {
    const int gid  = blockIdx.x * blockDim.x + threadIdx.x;
    const int e    = gid >> 5;
    const int lane = gid & 31;
    if (e >= E) return;

    const int src = ei[e];
    const int tgt = ei[E + e];
    const int rt  = et[e];

    const float4 hs = *(const float4*)(x     + (size_t)src * D + lane * 4);
    const float4 hr = *(const float4*)(rel   + (size_t)rt  * D + lane * 4);
    const float4 wa = *(const float4*)(wattn + lane * 4);

    float p = (hs.x + hr.x) * wa.x + (hs.y + hr.y) * wa.y +
              (hs.z + hr.z) * wa.z + (hs.w + hr.w) * wa.w;
#pragma unroll
    for (int off = 16; off > 0; off >>= 1)   // wave32 butterfly reduction
        p += __shfl_xor(p, off, 32);

    const float attn = 1.0f / (1.0f + __expf(-p));

    const float4 m = *(const float4*)(xt + (size_t)src * D + lane * 4);
    float* o = out + (size_t)tgt * D + lane * 4;
    atomicAdd(o + 0, m.x * attn);
    atomicAdd(o + 1, m.y * attn);
    atomicAdd(o + 2, m.z * attn);
    atomicAdd(o + 3, m.w * attn);
}

// ---------------------------------------------------------------------------
// Kernel 3: in-place vectorized ReLU
// ---------------------------------------------------------------------------
__global__ __launch_bounds__(256)
void rgcn_relu(float4* __restrict__ out, int n4) {
    int i = blockIdx.x * blockDim.x + threadIdx.x;
    if (i < n4) {
        float4 v = out[i];
        v.x = fmaxf(v.x, 0.0f);
        v.y = fmaxf(v.y, 0.0f);
        v.z = fmaxf(v.z, 0.0f);
        v.w = fmaxf(v.w, 0.0f);
        out[i] = v;
    }
}

extern "C" void kernel_launch(void* const* d_in, const int* in_sizes, int n_in,
                              void* d_out, int out_size, void* d_ws, size_t ws_size,
                              hipStream_t stream) {
    const float* x   = (const float*)d_in[0];  // [N, D]
    const int*   ei  = (const int*)  d_in[1];  // [2, E]
    const int*   et  = (const int*)  d_in[2];  // [E]
    const float* rel = (const float*)d_in[3];  // [R, D]
    const float* wl  = (const float*)d_in[4];  // [D, D] (out,in)
    const float* wa  = (const float*)d_in[5];  // [1, D]
    float* out = (float*)d_out;                // [N, D]
    float* xt  = (float*)d_ws;                 // scratch: x_trans [N, D] fp32

    const int N = in_sizes[0] / D;   // 50000
    const int E = in_sizes[2];       // 640000

    // atomics accumulate into out -> must start from zero every call
    hipMemsetAsync(d_out, 0, (size_t)out_size * sizeof(float), stream);

    const int mblocks = (N + 15) / 16;
    rgcn_lin_wmma<<<mblocks, 256, 0, stream>>>(x, wl, xt, N);

    const int eblocks = (E + 7) / 8;  // 8 waves (edges) per 256-thread block
    rgcn_edge<<<eblocks, 256, 0, stream>>>(x, ei, et, rel, wa, xt, out, E);

    const int n4 = out_size / 4;
    rgcn_relu<<<(n4 + 255) / 256, 256, 0, stream>>>((float4*)out, n4);
}